// DifferentiablePersistence_18983755448639
// MI455X (gfx1250) — compile-verified
//
#include <hip/hip_runtime.h>
#include <hip/hip_bf16.h>

#define N 768
#define NT 50
#define SIGMA 0.1f
#define RES 100
#define NL 5
#define TAYLOR_M 10
#define SQ_STEPS 16
#define THETA 0.5f

#define BM 128
#define BN 128
#define BK 16
#define LDA 20   // As row stride (M-major), even -> aligned float2 frags
#define LDBT 18  // Bs row stride (N-major, transposed), even -> aligned float2 frags

typedef __attribute__((ext_vector_type(2))) float v2f;
typedef __attribute__((ext_vector_type(8))) float v8f;

__device__ __forceinline__ float sigmoidf_(float x) {
  return 1.0f / (1.0f + __expf(-x));
}

// ---------------- pairwise distances ----------------
__global__ void dp_dist_kernel(const float* __restrict__ pts, float* __restrict__ dist) {
  int idx = blockIdx.x * blockDim.x + threadIdx.x;
  if (idx >= N * N) return;
  int i = idx / N, j = idx % N;
  float dx = pts[i * 3 + 0] - pts[j * 3 + 0];
  float dy = pts[i * 3 + 1] - pts[j * 3 + 1];
  float dz = pts[i * 3 + 2] - pts[j * 3 + 2];
  float d2 = dx * dx + dy * dy + dz * dz;
  dist[idx] = d2 > 0.0f ? sqrtf(d2) : 0.0f;
}

// ---------------- max dist + thresholds (also zero gershgorin accumulators) --------
__global__ void dp_maxthr_kernel(const float* __restrict__ dist, float* __restrict__ maxd,
                                 float* __restrict__ thr, float* __restrict__ gersh) {
  __shared__ float red[1024];
  int tid = threadIdx.x;
  float m = 0.0f;
  for (int i = tid; i < N * N; i += 1024) m = fmaxf(m, dist[i]);
  red[tid] = m;
  __syncthreads();
  for (int s = 512; s > 0; s >>= 1) {
    if (tid < s) red[tid] = fmaxf(red[tid], red[tid + s]);
    __syncthreads();
  }
  if (tid == 0) maxd[0] = red[0];
  __syncthreads();
  float md = red[0];
  if (tid < NT) {
    thr[tid] = md * (float)tid / (float)(NT - 1);
    gersh[tid] = 0.0f;
  }
}

// ---------------- degree rows + Gershgorin bound per threshold ----------------
__global__ void dp_deg_kernel(const float* __restrict__ dist, const float* __restrict__ thr,
                              float* __restrict__ deg, float* __restrict__ gersh) {
  int t = blockIdx.x / N;
  int i = blockIdx.x % N;
  int tid = threadIdx.x;
  float th = thr[t];
  float s = 0.0f;
  for (int j = tid; j < N; j += 256) s += sigmoidf_((th - dist[i * N + j]) * (1.0f / SIGMA));
  __shared__ float red[256];
  red[tid] = s;
  __syncthreads();
  for (int k = 128; k > 0; k >>= 1) {
    if (tid < k) red[tid] += red[tid + k];
    __syncthreads();
  }
  if (tid == 0) {
    float d = red[0];
    deg[t * N + i] = d;
    float aii = sigmoidf_(th * (1.0f / SIGMA));  // dist_ii == 0
    float g = 2.0f * (d - aii);                  // >= 0
    atomicMax((int*)&gersh[t], __float_as_int(g));
  }
}

// ---------------- per-threshold scaling exponent ----------------
__global__ void dp_prep_kernel(const float* __restrict__ gersh, float* __restrict__ invc,
                               float* __restrict__ sarr) {
  int t = threadIdx.x;
  if (t >= NT) return;
  float lam = gersh[t];
  float x = lam / (SIGMA * THETA);
  int s = 0;
  float p = 1.0f;
  while (p < x && s < SQ_STEPS) { p *= 2.0f; s++; }
  sarr[t] = (float)s;
  invc[t] = 1.0f / (SIGMA * p);  // M = (A - D) * invc has spectrum in [-THETA, 0]
}

// ---------------- build M and Horner seed B = I + M/m ----------------
__global__ void dp_buildm_kernel(const float* __restrict__ dist, const float* __restrict__ thr,
                                 const float* __restrict__ deg, const float* __restrict__ invc,
                                 float* __restrict__ M, float* __restrict__ B0, int t0) {
  int z = blockIdx.y;
  int t = t0 + z;
  int idx = blockIdx.x * 256 + threadIdx.x;
  int i = idx / N, j = idx % N;
  float a = sigmoidf_((thr[t] - dist[idx]) * (1.0f / SIGMA));
  float m = a;
  if (i == j) m -= deg[t * N + i];
  m *= invc[t];
  size_t o = (size_t)z * ((size_t)N * N) + idx;
  M[o] = m;
  B0[o] = m * (1.0f / (float)TAYLOR_M) + ((i == j) ? 1.0f : 0.0f);
}

// ---------------- batched f32 WMMA GEMM: D = scale*(A*B) [+ I] (or tile copy) -----
__global__ void __launch_bounds__(256)
dp_gemm_kernel(const float* __restrict__ Ab, const float* __restrict__ Bb,
               float* __restrict__ Db, float scale, int addI,
               const float* __restrict__ sarr, int step, int t0) {
  const int z = blockIdx.z;
  const size_t moff = (size_t)z * (size_t)N * N;
  const float* A = Ab + moff;
  const float* B = Bb + moff;
  float* D = Db + moff;
  const int RM = blockIdx.y * BM;
  const int CN = blockIdx.x * BN;
  const int tid = threadIdx.x;

  if (sarr) {
    int st = (int)sarr[t0 + z];
    if (step > st) {  // this threshold finished its squarings: pass-through copy
      for (int it = 0; it < 16; ++it) {
        int r = it * 8 + (tid >> 5);
        int c = (tid & 31) * 4;
        float4 v = *(const float4*)(B + (size_t)(RM + r) * N + CN + c);
        *(float4*)(D + (size_t)(RM + r) * N + CN + c) = v;
      }
      return;
    }
  }

  __shared__ float As[BM * LDA];   // [m][k]  m-major
  __shared__ float Bs[BN * LDBT];  // [n][k]  n-major (transposed during fill)

  const int lane = tid & 31;
  const int wid = tid >> 5;
  const int wm = wid >> 2;   // 0..1 -> 64-row slab
  const int wn = wid & 3;    // 0..3 -> 32-col slab
  const int half = lane >> 4;
  const int l16 = lane & 15;

  v8f zero = {};
  v8f acc[4][2];
  for (int mt = 0; mt < 4; ++mt)
    for (int nt = 0; nt < 2; ++nt) acc[mt][nt] = zero;

  for (int kb = 0; kb < N; kb += BK) {
    float4 aReg0, aReg1, bReg0, bReg1;
    {
      int ar = tid >> 2;        // 0..63
      int ac = (tid & 3) * 4;   // 0,4,8,12
      aReg0 = *(const float4*)(A + (size_t)(RM + ar) * N + kb + ac);
      aReg1 = *(const float4*)(A + (size_t)(RM + 64 + ar) * N + kb + ac);
      int br = tid >> 5;        // 0..7   (k within chunk)
      int bc = (tid & 31) * 4;  // 0..124 (n)
      bReg0 = *(const float4*)(B + (size_t)(kb + br) * N + CN + bc);
      bReg1 = *(const float4*)(B + (size_t)(kb + 8 + br) * N + CN + bc);
    }
    __syncthreads();  // previous iteration's WMMA reads done
    {
      int ar = tid >> 2;
      int ac = (tid & 3) * 4;
      *(float4*)(As + ar * LDA + ac) = aReg0;
      *(float4*)(As + (64 + ar) * LDA + ac) = aReg1;
      int br = tid >> 5;
      int bc = (tid & 31) * 4;
      // transpose into Bs[n][k]
      Bs[(bc + 0) * LDBT + br] = bReg0.x;
      Bs[(bc + 1) * LDBT + br] = bReg0.y;
      Bs[(bc + 2) * LDBT + br] = bReg0.z;
      Bs[(bc + 3) * LDBT + br] = bReg0.w;
      Bs[(bc + 0) * LDBT + 8 + br] = bReg1.x;
      Bs[(bc + 1) * LDBT + 8 + br] = bReg1.y;
      Bs[(bc + 2) * LDBT + 8 + br] = bReg1.z;
      Bs[(bc + 3) * LDBT + 8 + br] = bReg1.w;
    }
    __syncthreads();

    for (int kk = 0; kk < 4; ++kk) {
      int k0 = kk * 4 + half * 2;  // lanes 0-15: K={k0,k0+1}; lanes 16-31: K+2
      v2f afrag[4], bfrag[2];
      for (int mt = 0; mt < 4; ++mt) {
        int m = wm * 64 + mt * 16 + l16;
        afrag[mt] = *(const v2f*)(As + m * LDA + k0);
      }
      for (int nt = 0; nt < 2; ++nt) {
        int n = wn * 32 + nt * 16 + l16;
        bfrag[nt] = *(const v2f*)(Bs + n * LDBT + k0);
      }
      for (int mt = 0; mt < 4; ++mt)
        for (int nt = 0; nt < 2; ++nt)
          acc[mt][nt] = __builtin_amdgcn_wmma_f32_16x16x4_f32(
              false, afrag[mt], false, bfrag[nt], (short)0, acc[mt][nt], false, false);
    }
  }

  for (int mt = 0; mt < 4; ++mt) {
    for (int nt = 0; nt < 2; ++nt) {
      int col = CN + wn * 32 + nt * 16 + l16;
      for (int r = 0; r < 8; ++r) {
        int row = RM + wm * 64 + mt * 16 + half * 8 + r;
        float v = acc[mt][nt][r] * scale;
        if (addI && row == col) v += 1.0f;
        D[(size_t)row * N + col] = v;
      }
    }
  }
}

// ---------------- trace ----------------
__global__ void dp_trace_kernel(const float* __restrict__ P, float* __restrict__ betti, int t0) {
  int z = blockIdx.x;
  int tid = threadIdx.x;
  float s = 0.0f;
  for (int i = tid; i < N; i += 256) s += P[(size_t)z * N * N + (size_t)i * N + i];
  __shared__ float red[256];
  red[tid] = s;
  __syncthreads();
  for (int k = 128; k > 0; k >>= 1) {
    if (tid < k) red[tid] += red[tid + k];
    __syncthreads();
  }
  if (tid == 0) betti[t0 + z] = red[0];
}

// ---------------- interpolation + landscapes ----------------
__global__ void dp_landscape_kernel(const float* __restrict__ betti, float* __restrict__ out) {
  __shared__ float bi[RES];
  __shared__ float sm[RES];
  __shared__ float dv[RES];
  __shared__ float stat[2];
  int r = threadIdx.x;
  if (r < RES) {
    float pos = (float)r * (float)(NT - 1) / (float)(RES - 1);
    int i0 = (int)floorf(pos);
    if (i0 > NT - 2) i0 = NT - 2;
    if (i0 < 0) i0 = 0;
    float frac = pos - (float)i0;
    bi[r] = betti[i0] * (1.0f - frac) + betti[i0 + 1] * frac;
  }
  __syncthreads();
  if (r == 0) {
    float mx = bi[0];
    for (int q = 1; q < RES; ++q) mx = fmaxf(mx, bi[q]);
    stat[0] = mx + 1e-8f;
  }
  __syncthreads();
  if (r < RES) out[r] = bi[r] / stat[0];

  for (int k = 1; k < NL; ++k) {
    int ks = 2 * k + 1;  // 3,5,7,9 (all < RES/4 = 25)
    int pad = ks / 2;
    __syncthreads();
    if (r < RES) {
      float s = 0.0f;
      for (int u = 0; u < ks; ++u) {
        int j = r - pad + u;
        if (j < 0) j = 0;
        if (j > RES - 1) j = RES - 1;
        s += bi[j];
      }
      sm[r] = s / (float)ks;
    }
    __syncthreads();
    if (r < RES - 1) dv[r] = sm[r + 1] - sm[r];
    __syncthreads();
    if (r == 0) {
      dv[RES - 1] = dv[RES - 2];
      float mx = 0.0f;
      for (int q = 0; q < RES; ++q) mx = fmaxf(mx, fabsf(dv[q]));
      stat[1] = mx + 1e-8f;
    }
    __syncthreads();
    if (r < RES) out[k * RES + r] = dv[r] / stat[1];
  }
}

extern "C" void kernel_launch(void* const* d_in, const int* in_sizes, int n_in,
                              void* d_out, int out_size, void* d_ws, size_t ws_size,
                              hipStream_t stream) {
  (void)in_sizes; (void)n_in; (void)out_size;
  const float* pts = (const float*)d_in[0];
  float* out = (float*)d_out;
  float* w = (float*)d_ws;

  const size_t NNs = (size_t)N * N;
  float* dist  = w;                                   // NN
  float* maxd  = w + NNs;                             // 1
  float* thr   = w + NNs + 16;                        // NT
  float* deg   = w + NNs + 80;                        // NT*N
  float* gersh = w + NNs + 80 + (size_t)NT * N;       // 64
  float* invc  = gersh + 64;                          // 64
  float* sarr  = invc + 64;                           // 64
  float* betti = sarr + 64;                           // 64
  size_t head = NNs + 80 + (size_t)NT * N + 256;
  head = (head + 63) & ~(size_t)63;

  size_t availF = ws_size / 4;
  size_t bufF = availF > head ? availF - head : 0;
  int Tc = (int)(bufF / (3 * NNs));
  if (Tc > NT) Tc = NT;
  if (Tc < 1) Tc = 1;

  float* bufM = w + head;
  float* buf0 = bufM + (size_t)Tc * NNs;
  float* buf1 = buf0 + (size_t)Tc * NNs;

  dp_dist_kernel<<<(N * N + 255) / 256, 256, 0, stream>>>(pts, dist);
  dp_maxthr_kernel<<<1, 1024, 0, stream>>>(dist, maxd, thr, gersh);
  dp_deg_kernel<<<NT * N, 256, 0, stream>>>(dist, thr, deg, gersh);
  dp_prep_kernel<<<1, 64, 0, stream>>>(gersh, invc, sarr);

  for (int t0 = 0; t0 < NT; t0 += Tc) {
    int tc = (NT - t0) < Tc ? (NT - t0) : Tc;
    dim3 gb((N * N) / 256, tc);
    dp_buildm_kernel<<<gb, 256, 0, stream>>>(dist, thr, deg, invc, bufM, buf0, t0);

    dim3 gg(N / BN, N / BM, tc);
    float* src = buf0;
    float* dst = buf1;
    // Horner: B <- I + (M*B)/k, k = m-1 .. 1
    for (int k = TAYLOR_M - 1; k >= 1; --k) {
      dp_gemm_kernel<<<gg, 256, 0, stream>>>(bufM, src, dst, 1.0f / (float)k, 1, nullptr, 0, t0);
      float* tmp = src; src = dst; dst = tmp;
    }
    // squarings: P <- P*P (copy once past this threshold's s_t)
    for (int j = 1; j <= SQ_STEPS; ++j) {
      dp_gemm_kernel<<<gg, 256, 0, stream>>>(src, src, dst, 1.0f, 0, sarr, j, t0);
      float* tmp = src; src = dst; dst = tmp;
    }
    dp_trace_kernel<<<tc, 256, 0, stream>>>(src, betti, t0);
  }

  dp_landscape_kernel<<<1, 128, 0, stream>>>(betti, out);
}